// cross_att_fusion_block_2_83434034692675
// MI455X (gfx1250) — compile-verified
//
#include <hip/hip_runtime.h>

typedef __attribute__((ext_vector_type(2))) float v2f;
typedef __attribute__((ext_vector_type(8))) float v8f;

#define NP   50000      // points
#define EPSV 1e-5f

// Parameter order inside each _make_params() dict (26 arrays per set)
enum {
  P_Wq = 0, P_bq, P_Wk, P_bk, P_Wv, P_bv,
  P_p1w, P_p1b, P_pbng, P_pbnb, P_pbnm, P_pbnv,
  P_p2w, P_p2b,
  P_wbn1g, P_wbn1b, P_wbn1m, P_wbn1v,
  P_w1w, P_w1b,
  P_wbn2g, P_wbn2b, P_wbn2m, P_wbn2v,
  P_w2w, P_w2b,
  P_COUNT
};

// ---------------------------------------------------------------------------
// Kernel A: 4 dense GEMMs  Y[64,N] = W[64,64] * X[64,N] + bias
//   g=0: Kf0 = Wk_spa * spe ; g=1: V0 = Wv_spa * spa
//   g=2: Kf1 = Wk_spe * spa ; g=3: V1 = Wv_spe * spe
// One wave computes a 64x16 tile via 16 K-steps of V_WMMA_F32_16X16X4_F32.
// ---------------------------------------------------------------------------
struct GemmArgs {
  const float* W[4];
  const float* bias[4];
  const float* X[4];
  float*       Y[4];
};

__global__ __launch_bounds__(128)
void qkv_gemm(GemmArgs ga) {
  const int g = blockIdx.y;
  const float* __restrict__ W    = ga.W[g];
  const float* __restrict__ bias = ga.bias[g];
  const float* __restrict__ X    = ga.X[g];
  float* __restrict__       Y    = ga.Y[g];

  const int wave = threadIdx.x >> 5;
  const int lane = threadIdx.x & 31;
  const int tile = blockIdx.x * 4 + wave;
  if (tile >= NP / 16) return;
  const int n0   = tile * 16;
  const int half = lane >> 4;     // 0: K={0,1}, 1: K={2,3} within a K-step
  const int m    = lane & 15;     // A row / B column

  v8f acc0 = {0.f,0.f,0.f,0.f,0.f,0.f,0.f,0.f};
  v8f acc1 = acc0, acc2 = acc0, acc3 = acc0;

  #pragma unroll
  for (int k = 0; k < 16; ++k) {
    const int cb = k * 4 + 2 * half;
    v2f b;
    b.x = X[cb * NP + n0 + m];
    b.y = X[(cb + 1) * NP + n0 + m];
    v2f a;
    a.x = W[m * 64 + cb];            a.y = W[m * 64 + cb + 1];
    acc0 = __builtin_amdgcn_wmma_f32_16x16x4_f32(false, a, false, b, (short)0, acc0, false, false);
    a.x = W[(16 + m) * 64 + cb];     a.y = W[(16 + m) * 64 + cb + 1];
    acc1 = __builtin_amdgcn_wmma_f32_16x16x4_f32(false, a, false, b, (short)0, acc1, false, false);
    a.x = W[(32 + m) * 64 + cb];     a.y = W[(32 + m) * 64 + cb + 1];
    acc2 = __builtin_amdgcn_wmma_f32_16x16x4_f32(false, a, false, b, (short)0, acc2, false, false);
    a.x = W[(48 + m) * 64 + cb];     a.y = W[(48 + m) * 64 + cb + 1];
    acc3 = __builtin_amdgcn_wmma_f32_16x16x4_f32(false, a, false, b, (short)0, acc3, false, false);
  }

  // D layout: VGPR r holds row (r + 8*half), column = lane&15
  #pragma unroll
  for (int r = 0; r < 8; ++r) {
    const int row = r + 8 * half;
    Y[row * NP        + n0 + m] = acc0[r] + bias[row];
    Y[(16 + row) * NP + n0 + m] = acc1[r] + bias[16 + row];
    Y[(32 + row) * NP + n0 + m] = acc2[r] + bias[32 + row];
    Y[(48 + row) * NP + n0 + m] = acc3[r] + bias[48 + row];
  }
}

// ---------------------------------------------------------------------------
// Kernel B: fused per-16-point attention block (one wave per block)
// ---------------------------------------------------------------------------
struct PTArgs {
  const float *Wq, *bq;
  const float *p1w, *p1b, *pbng, *pbnb, *pbnm, *pbnv, *p2w, *p2b;
  const float *bn1g, *bn1b, *bn1m, *bn1v, *w1w, *w1b;
  const float *bn2g, *bn2b, *bn2m, *bn2v, *w2w, *w2b;
  const float *qin, *Kf, *Vf;   // qin = query/residual input [64,N]
  float*       outb;            // d_out + layer*64*N
};

__global__ __launch_bounds__(32)
void pt_block(PTArgs A0, PTArgs A1,
              const float* __restrict__ coord,
              const int* __restrict__ nidx) {
  const PTArgs A  = (blockIdx.y == 0) ? A0 : A1;
  const int lane  = threadIdx.x;
  const int p0    = blockIdx.x * 16;

  __shared__ float xq[64][16];     // x_q tile
  __shared__ float vv[256 * 3];    // relu(bn(conv3x3(coord_nbr)))
  __shared__ float lg[256 * 4];    // logits -> softmax weights
  __shared__ int   nb_s[256];      // neighbor indices
  __shared__ float s1[64], t1[64]; // fused BN1 scale/shift

  // neighbor indices (coalesced)
  #pragma unroll
  for (int i = 0; i < 8; ++i)
    nb_s[i * 32 + lane] = nidx[p0 * 16 + i * 32 + lane];

  // fused BN1 constants: h = max(0, w*s1 + t1)
  #pragma unroll
  for (int j = 0; j < 2; ++j) {
    const int c = lane + j * 32;
    const float sc = A.bn1g[c] * rsqrtf(A.bn1v[c] + EPSV);
    s1[c] = sc;
    t1[c] = A.bn1b[c] - A.bn1m[c] * sc;
  }

  // x_q = Wq * qin + bq for the 16 points of this block (WMMA)
  {
    const int half = lane >> 4;
    const int m    = lane & 15;
    v8f acc0 = {0.f,0.f,0.f,0.f,0.f,0.f,0.f,0.f};
    v8f acc1 = acc0, acc2 = acc0, acc3 = acc0;
    #pragma unroll
    for (int k = 0; k < 16; ++k) {
      const int cb = k * 4 + 2 * half;
      v2f b;
      b.x = A.qin[cb * NP + p0 + m];
      b.y = A.qin[(cb + 1) * NP + p0 + m];
      v2f a;
      a.x = A.Wq[m * 64 + cb];          a.y = A.Wq[m * 64 + cb + 1];
      acc0 = __builtin_amdgcn_wmma_f32_16x16x4_f32(false, a, false, b, (short)0, acc0, false, false);
      a.x = A.Wq[(16 + m) * 64 + cb];   a.y = A.Wq[(16 + m) * 64 + cb + 1];
      acc1 = __builtin_amdgcn_wmma_f32_16x16x4_f32(false, a, false, b, (short)0, acc1, false, false);
      a.x = A.Wq[(32 + m) * 64 + cb];   a.y = A.Wq[(32 + m) * 64 + cb + 1];
      acc2 = __builtin_amdgcn_wmma_f32_16x16x4_f32(false, a, false, b, (short)0, acc2, false, false);
      a.x = A.Wq[(48 + m) * 64 + cb];   a.y = A.Wq[(48 + m) * 64 + cb + 1];
      acc3 = __builtin_amdgcn_wmma_f32_16x16x4_f32(false, a, false, b, (short)0, acc3, false, false);
    }
    #pragma unroll
    for (int r = 0; r < 8; ++r) {
      const int row = r + 8 * half;
      xq[row][m]      = acc0[r] + A.bq[row];
      xq[16 + row][m] = acc1[r] + A.bq[16 + row];
      xq[32 + row][m] = acc2[r] + A.bq[32 + row];
      xq[48 + row][m] = acc3[r] + A.bq[48 + row];
    }
  }
  __syncthreads();

  // Pass 1: per (pt,nb) pair -> positional vec + attention logits
  #pragma unroll
  for (int i = 0; i < 8; ++i) {
    const int pair = i * 32 + lane;
    const int pt   = pair >> 4;
    const int nbr  = nb_s[pair];
    const float c0 = coord[nbr * 3 + 0];
    const float c1 = coord[nbr * 3 + 1];
    const float c2 = coord[nbr * 3 + 2];
    float v[3];
    #pragma unroll
    for (int o = 0; o < 3; ++o) {
      float t  = A.p1w[o*3+0]*c0 + A.p1w[o*3+1]*c1 + A.p1w[o*3+2]*c2 + A.p1b[o];
      float sc = A.pbng[o] * rsqrtf(A.pbnv[o] + EPSV);
      t = (t - A.pbnm[o]) * sc + A.pbnb[o];
      v[o] = fmaxf(t, 0.f);
      vv[pair * 3 + o] = v[o];
    }
    float a0 = 0.f, a1 = 0.f, a2 = 0.f, a3 = 0.f;
    const float* kcol = A.Kf + nbr;
    #pragma unroll 8
    for (int c = 0; c < 64; ++c) {
      const float pr = A.p2w[c*3+0]*v[0] + A.p2w[c*3+1]*v[1] + A.p2w[c*3+2]*v[2] + A.p2b[c];
      const float wv = kcol[c * NP] - xq[c][pt] + pr;
      const float h  = fmaxf(fmaf(wv, s1[c], t1[c]), 0.f);
      a0 = fmaf(A.w1w[c],       h, a0);
      a1 = fmaf(A.w1w[64 + c],  h, a1);
      a2 = fmaf(A.w1w[128 + c], h, a2);
      a3 = fmaf(A.w1w[192 + c], h, a3);
    }
    const float aa[4] = {a0, a1, a2, a3};
    float h2[4];
    #pragma unroll
    for (int j = 0; j < 4; ++j) {
      const float sc = A.bn2g[j] * rsqrtf(A.bn2v[j] + EPSV);
      const float t  = ((aa[j] + A.w1b[j]) - A.bn2m[j]) * sc + A.bn2b[j];
      h2[j] = fmaxf(t, 0.f);
    }
    #pragma unroll
    for (int cs = 0; cs < 4; ++cs)
      lg[pair * 4 + cs] = A.w2w[cs*4+0]*h2[0] + A.w2w[cs*4+1]*h2[1]
                        + A.w2w[cs*4+2]*h2[2] + A.w2w[cs*4+3]*h2[3] + A.w2b[cs];
  }
  __syncthreads();

  // Pass 2: softmax over the 16 neighbors, per (pt, cs) row (64 rows, 2/lane)
  #pragma unroll
  for (int rr = 0; rr < 2; ++rr) {
    const int row  = lane * 2 + rr;
    const int pt   = row >> 2;
    const int cs   = row & 3;
    const int base = pt * 64 + cs;        // (pt*16+nb)*4 + cs
    float mx = -3.4e38f;
    #pragma unroll
    for (int nb = 0; nb < 16; ++nb) mx = fmaxf(mx, lg[base + nb * 4]);
    float s = 0.f;
    #pragma unroll
    for (int nb = 0; nb < 16; ++nb) {
      const float e = __expf(lg[base + nb * 4] - mx);
      lg[base + nb * 4] = e;
      s += e;
    }
    const float inv = 1.f / s;
    #pragma unroll
    for (int nb = 0; nb < 16; ++nb) lg[base + nb * 4] *= inv;
  }
  __syncthreads();

  // Pass 3: out[c,pt] = qin[c,pt] + sum_nb (V[c,nbr] + p_r) * soft[pt,nb,c&3]
  for (int i = 0; i < 32; ++i) {
    const int pr2 = i * 32 + lane;
    const int c   = pr2 >> 4;
    const int pt  = pr2 & 15;
    const int cs  = c & 3;
    const float pw0 = A.p2w[c*3+0], pw1 = A.p2w[c*3+1], pw2 = A.p2w[c*3+2];
    const float pb  = A.p2b[c];
    const float* vcol = A.Vf + c * NP;
    float acc = 0.f;
    #pragma unroll
    for (int nb = 0; nb < 16; ++nb) {
      const int q   = pt * 16 + nb;
      const int nbr = nb_s[q];
      const float pr = pw0 * vv[q*3+0] + pw1 * vv[q*3+1] + pw2 * vv[q*3+2] + pb;
      acc = fmaf(vcol[nbr] + pr, lg[q * 4 + cs], acc);
    }
    const int col = p0 + pt;
    A.outb[c * NP + col] = A.qin[c * NP + col] + acc;
  }
}

// ---------------------------------------------------------------------------
extern "C" void kernel_launch(void* const* d_in, const int* in_sizes, int n_in,
                              void* d_out, int out_size, void* d_ws, size_t ws_size,
                              hipStream_t stream) {
  (void)in_sizes; (void)n_in; (void)out_size; (void)ws_size;

  const float* coord = (const float*)d_in[0];
  const float* spa   = (const float*)d_in[1];
  const float* spe   = (const float*)d_in[2];
  const int*   nidx  = (const int*)d_in[3];

  const float* P0[P_COUNT];
  const float* P1[P_COUNT];
  for (int j = 0; j < P_COUNT; ++j) {
    P0[j] = (const float*)d_in[4 + j];
    P1[j] = (const float*)d_in[4 + P_COUNT + j];
  }

  float* out = (float*)d_out;
  float* ws  = (float*)d_ws;
  float* Kf0 = ws;
  float* V0  = ws + (size_t)64 * NP;
  float* Kf1 = ws + (size_t)2 * 64 * NP;
  float* V1  = ws + (size_t)3 * 64 * NP;

  GemmArgs ga;
  ga.W[0] = P0[P_Wk]; ga.bias[0] = P0[P_bk]; ga.X[0] = spe; ga.Y[0] = Kf0;
  ga.W[1] = P0[P_Wv]; ga.bias[1] = P0[P_bv]; ga.X[1] = spa; ga.Y[1] = V0;
  ga.W[2] = P1[P_Wk]; ga.bias[2] = P1[P_bk]; ga.X[2] = spa; ga.Y[2] = Kf1;
  ga.W[3] = P1[P_Wv]; ga.bias[3] = P1[P_bv]; ga.X[3] = spe; ga.Y[3] = V1;
  qkv_gemm<<<dim3((NP / 16 + 3) / 4, 4), 128, 0, stream>>>(ga);

  auto fill = [](const float* const* P, const float* qin,
                 const float* Kf, const float* Vf, float* outb) {
    PTArgs a;
    a.Wq   = P[P_Wq];    a.bq   = P[P_bq];
    a.p1w  = P[P_p1w];   a.p1b  = P[P_p1b];
    a.pbng = P[P_pbng];  a.pbnb = P[P_pbnb];
    a.pbnm = P[P_pbnm];  a.pbnv = P[P_pbnv];
    a.p2w  = P[P_p2w];   a.p2b  = P[P_p2b];
    a.bn1g = P[P_wbn1g]; a.bn1b = P[P_wbn1b];
    a.bn1m = P[P_wbn1m]; a.bn1v = P[P_wbn1v];
    a.w1w  = P[P_w1w];   a.w1b  = P[P_w1b];
    a.bn2g = P[P_wbn2g]; a.bn2b = P[P_wbn2b];
    a.bn2m = P[P_wbn2m]; a.bn2v = P[P_wbn2v];
    a.w2w  = P[P_w2w];   a.w2b  = P[P_w2b];
    a.qin  = qin; a.Kf = Kf; a.Vf = Vf; a.outb = outb;
    return a;
  };

  PTArgs a0 = fill(P0, spa, Kf0, V0, out);
  PTArgs a1 = fill(P1, spe, Kf1, V1, out + (size_t)64 * NP);
  pt_block<<<dim3(NP / 16, 2), 32, 0, stream>>>(a0, a1, coord, nidx);
}